// GCN_11364483465281
// MI455X (gfx1250) — compile-verified
//
#include <hip/hip_runtime.h>
#include <hip/hip_bf16.h>

#define N_NODES  50000
#define N_EDGES  800000
#define FEATS    128      // in-feats == hidden width
#define CLASSES  64

typedef float v2f __attribute__((ext_vector_type(2)));
typedef float v8f __attribute__((ext_vector_type(8)));

// ---------------------------------------------------------------- utilities
__global__ void zero_kernel(float* __restrict__ p, long n) {
    long i = (long)blockIdx.x * blockDim.x + threadIdx.x;
    if (i < n) p[i] = 0.0f;
}

// degrees accumulated as float (values are small integers, exact in f32)
__global__ void degree_kernel(const int* __restrict__ src, const int* __restrict__ dst,
                              float* __restrict__ odeg, float* __restrict__ ideg, int ne) {
    int e = blockIdx.x * blockDim.x + threadIdx.x;
    if (e >= ne) return;
    atomicAdd(&odeg[src[e]], 1.0f);
    atomicAdd(&ideg[dst[e]], 1.0f);
}

// deg -> clip(deg,1)^-0.5  (in place, both arrays)
__global__ void norm_kernel(float* __restrict__ a, float* __restrict__ b, int n) {
    int i = blockIdx.x * blockDim.x + threadIdx.x;
    if (i >= n) return;
    a[i] = rsqrtf(fmaxf(a[i], 1.0f));
    b[i] = rsqrtf(fmaxf(b[i], 1.0f));
}

// ------------------------------------------------- edge gather + scatter-add
// agg[dst] += h[src] * norm_src[src].  One wave handles one edge's 128-f32 row
// (each lane: one float4 => coalesced 512B row gather + 4 f32 atomics).
__global__ void scatter_kernel(const float* __restrict__ h,
                               const float* __restrict__ nsrc,
                               const int* __restrict__ src, const int* __restrict__ dst,
                               float* __restrict__ agg, int ne) {
    long t = (long)blockIdx.x * blockDim.x + threadIdx.x;
    int e = (int)(t >> 5);
    if (e >= ne) return;
    int c = (int)(t & 31) << 2;                 // feature offset 0..124
    int s = src[e], d = dst[e];
    float ns = nsrc[s];
    const float4 v = *reinterpret_cast<const float4*>(h + (size_t)s * FEATS + c);
    float* dp = agg + (size_t)d * FEATS + c;
    atomicAdd(dp + 0, v.x * ns);
    atomicAdd(dp + 1, v.y * ns);
    atomicAdd(dp + 2, v.z * ns);
    atomicAdd(dp + 3, v.w * ns);
}

// ------------------------------------------------------------- WMMA f32 GEMM
// out[M x N] = relu?( (agg * norm_dst) @ W + b ), K = 128 fixed.
// Block = 256 threads = 8 waves. Block owns one 16-row M tile (staged in LDS,
// pre-scaled by norm_dst); each wave owns one 16-col N tile and runs 32
// chained V_WMMA_F32_16X16X4_F32 ops over K.
// ISA 7.12.2 wave32 layouts: A(m=lane%16, k=2*(lane>>4)+v),
//                            B(k=2*(lane>>4)+v, n=lane%16),
//                            C/D(m=v+8*(lane>>4), n=lane%16).
__global__ __launch_bounds__(256) void gemm_wmma_kernel(
        const float* __restrict__ agg, const float* __restrict__ ndst,
        const float* __restrict__ W,   const float* __restrict__ bias,
        float* __restrict__ out, int N, int ldout, int relu) {
    __shared__ float Atile[16 * FEATS];         // 8 KB of the 320 KB WGP LDS
    const int tid  = threadIdx.x;
    const int row0 = blockIdx.x << 4;           // M tile base (M % 16 == 0)

    for (int i = tid; i < 16 * FEATS; i += 256) {
        int r = i >> 7, c = i & 127;
        Atile[i] = agg[(size_t)(row0 + r) * FEATS + c] * ndst[row0 + r];
    }
    __syncthreads();

    const int wave   = tid >> 5;
    const int lane   = tid & 31;
    const int ntiles = N >> 4;                  // 8 (hidden) or 4 (classes)
    if (wave >= ntiles) return;                 // whole-wave exit: EXEC stays all-1s for WMMA
    const int col0  = wave << 4;
    const int nrow  = lane & 15;                // m / n index within tile
    const int khalf = lane >> 4;                // K sub-phase select

    v8f acc = {0.f, 0.f, 0.f, 0.f, 0.f, 0.f, 0.f, 0.f};
    #pragma unroll 4
    for (int kb = 0; kb < FEATS; kb += 4) {
        const int k0 = kb + (khalf << 1);
        v2f a, b;
        a.x = Atile[nrow * FEATS + k0];
        a.y = Atile[nrow * FEATS + k0 + 1];
        b.x = W[(size_t)(k0 + 0) * N + col0 + nrow];
        b.y = W[(size_t)(k0 + 1) * N + col0 + nrow];
        // 8 args: (neg_a, A, neg_b, B, c_mod, C, reuse_a, reuse_b)
        acc = __builtin_amdgcn_wmma_f32_16x16x4_f32(
                false, a, false, b, (short)0, acc, false, false);
    }

    const float bb = bias[col0 + nrow];
    #pragma unroll
    for (int v = 0; v < 8; ++v) {
        float x = acc[v] + bb;
        if (relu) x = fmaxf(x, 0.0f);
        out[(size_t)(row0 + v + (khalf << 3)) * ldout + col0 + nrow] = x;
    }
}

// ---------------------------------------------------------------- read-outs
// mN[c] = sum_n h[n][c]   (h is M x 64); per-thread strided partial + atomic
__global__ void colsum_kernel(const float* __restrict__ h, float* __restrict__ mN, int M) {
    int g = blockIdx.x * blockDim.x + threadIdx.x;
    int c = g & 63;
    int r0 = g >> 6;
    int stride = (int)((gridDim.x * blockDim.x) >> 6);
    float s = 0.0f;
    for (int n = r0; n < M; n += stride) s += h[(size_t)n * CLASSES + c];
    atomicAdd(&mN[c], s);
}

// PI[n] = h[n] . Wp + bp
__global__ void pi_kernel(const float* __restrict__ h, const float* __restrict__ Wp,
                          const float* __restrict__ bp, float* __restrict__ out, int M) {
    int n = blockIdx.x * blockDim.x + threadIdx.x;
    if (n >= M) return;
    const float* r = h + (size_t)n * CLASSES;
    float s = 0.0f;
    #pragma unroll
    for (int c = 0; c < CLASSES; ++c) s += r[c] * Wp[c];
    out[n] = s + bp[0];
}

// V = mN . Wv + bv  -> out[N_NODES]
__global__ void v_kernel(const float* __restrict__ mN, const float* __restrict__ Wv,
                         const float* __restrict__ bv, float* __restrict__ out) {
    if (threadIdx.x == 0 && blockIdx.x == 0) {
        float s = 0.0f;
        for (int c = 0; c < CLASSES; ++c) s += mN[c] * Wv[c];
        out[N_NODES] = s + bv[0];
    }
}

// ------------------------------------------------------------------- driver
extern "C" void kernel_launch(void* const* d_in, const int* in_sizes, int n_in,
                              void* d_out, int out_size, void* d_ws, size_t ws_size,
                              hipStream_t stream) {
    (void)in_sizes; (void)n_in; (void)out_size; (void)ws_size;
    const float* features = (const float*)d_in[0];
    const int*   src      = (const int*)d_in[1];
    const int*   dst      = (const int*)d_in[2];
    const float* W[4] = {(const float*)d_in[3], (const float*)d_in[5],
                         (const float*)d_in[7], (const float*)d_in[9]};
    const float* B[4] = {(const float*)d_in[4], (const float*)d_in[6],
                         (const float*)d_in[8], (const float*)d_in[10]};
    const float* Wp = (const float*)d_in[11];
    const float* bp = (const float*)d_in[12];
    const float* Wv = (const float*)d_in[13];
    const float* bv = (const float*)d_in[14];
    float* out = (float*)d_out;

    // workspace carve-up (floats): 2*50000 norms + 3*(50000*128) + 64 ~= 77 MB
    float* ws    = (float*)d_ws;
    float* nrm_s = ws;
    float* nrm_d = nrm_s + N_NODES;
    float* bufA  = nrm_d + N_NODES;
    float* bufB  = bufA + (size_t)N_NODES * FEATS;
    float* agg   = bufB + (size_t)N_NODES * FEATS;
    float* mN    = agg  + (size_t)N_NODES * FEATS;

    const int T = 256;

    // degrees -> symmetric norms (nrm_s/nrm_d are contiguous: zero both at once)
    zero_kernel<<<(2L * N_NODES + T - 1) / T, T, 0, stream>>>(nrm_s, 2L * N_NODES);
    degree_kernel<<<(N_EDGES + T - 1) / T, T, 0, stream>>>(src, dst, nrm_s, nrm_d, N_EDGES);
    norm_kernel<<<(N_NODES + T - 1) / T, T, 0, stream>>>(nrm_s, nrm_d, N_NODES);

    const long aggN     = (long)N_NODES * FEATS;
    const long sthreads = (long)N_EDGES * 32;

    const float* hin = features;
    float* hout = bufA;
    for (int l = 0; l < 4; ++l) {
        zero_kernel<<<(aggN + T - 1) / T, T, 0, stream>>>(agg, aggN);
        scatter_kernel<<<(sthreads + T - 1) / T, T, 0, stream>>>(hin, nrm_s, src, dst, agg, N_EDGES);
        const int Nw  = (l < 3) ? FEATS : CLASSES;
        gemm_wmma_kernel<<<N_NODES / 16, 256, 0, stream>>>(
            agg, nrm_d, W[l], B[l], hout, Nw, Nw, (l < 3) ? 1 : 0);
        hin  = hout;
        hout = (hout == bufA) ? bufB : bufA;
    }
    const float* hfin = hin;   // N_NODES x 64, ld = 64

    zero_kernel<<<1, 64, 0, stream>>>(mN, 64);
    colsum_kernel<<<256, 256, 0, stream>>>(hfin, mN, N_NODES);
    pi_kernel<<<(N_NODES + T - 1) / T, T, 0, stream>>>(hfin, Wp, bp, out, N_NODES);
    v_kernel<<<1, 1, 0, stream>>>(mN, Wv, bv, out);
}